// DimDynamicLinear_16612933501330
// MI455X (gfx1250) — compile-verified
//
#include <hip/hip_runtime.h>

typedef __attribute__((ext_vector_type(2))) float v2f;
typedef __attribute__((ext_vector_type(8))) float v8f;

#define BM   128   // output tile rows (tokens)
#define BN   128   // output tile cols (selected features)
#define KT   16    // K-slice per LDS stage
#define LPAD 20    // LDS row pitch in floats: conflict-free mod 64, 16B-aligned rows

// ---------------------------------------------------------------------------
// Kernel 1: compact mask -> (idx, bias_sel) in ascending index order.
// One workgroup of 1024 threads; out_f = 16384 -> 16 entries per thread.
// ---------------------------------------------------------------------------
__global__ void compact_mask_kernel(const int* __restrict__ mask,
                                    const float* __restrict__ bias,
                                    int* __restrict__ idx,
                                    float* __restrict__ bsel,
                                    int out_f) {
    __shared__ int s[1024];
    const int t   = threadIdx.x;
    const int per = out_f >> 10;          // out_f / 1024
    const int base = t * per;

    int cnt = 0;
    for (int j = 0; j < per; ++j) cnt += (mask[base + j] != 0);
    s[t] = cnt;
    __syncthreads();

    // Hillis-Steele inclusive scan over 1024 counts
    for (int off = 1; off < 1024; off <<= 1) {
        int v = (t >= off) ? s[t - off] : 0;
        __syncthreads();
        s[t] += v;
        __syncthreads();
    }

    int o = s[t] - cnt;                   // exclusive prefix
    for (int j = 0; j < per; ++j) {
        const int g = base + j;
        if (mask[g] != 0) {
            idx[o]  = g;
            bsel[o] = bias[g];
            ++o;
        }
    }
}

// ---------------------------------------------------------------------------
// Kernel 2: out[M,N] = data[M,KD] * weight[idx[N],KD]^T + bsel[N]
// 256 threads = 8 waves. Tile 128x128, K-sliced by 16 through LDS.
// Each wave: 64x32 slab = 4x2 subtiles of 16x16, f32 WMMA 16x16x4.
// Assumes M%128==0, N%128==0, KD%16==0 (true for this problem).
// ---------------------------------------------------------------------------
__global__ void __launch_bounds__(256)
gemm_sel_kernel(const float* __restrict__ data,    // [M, KD]
                const float* __restrict__ weight,  // [OUT_F, KD]
                const int*   __restrict__ idx,     // [N]
                const float* __restrict__ bsel,    // [N]
                float*       __restrict__ out,     // [M, N]
                int M, int N, int KD) {
    __shared__ float As[BM * LPAD];
    __shared__ float Bs[BN * LPAD];

    const int tid  = threadIdx.x;
    const int lane = tid & 31;
    const int wave = tid >> 5;
    const int wm   = wave & 1;            // 0..1 -> 64-row slab
    const int wn   = wave >> 1;           // 0..3 -> 32-col slab

    const int row0 = blockIdx.y * BM;
    const int col0 = blockIdx.x * BN;

    // Global->LDS loader mapping: 512 float4 per tile (A and B each),
    // 2 float4 per thread.  q -> (row = q>>2, quad = q&3)
    const int q0 = tid, q1 = tid + 256;
    const int r0 = q0 >> 2, c0 = (q0 & 3) * 4;
    const int r1 = q1 >> 2, c1 = (q1 & 3) * 4;

    const float* aptr0 = data + (size_t)(row0 + r0) * KD + c0;
    const float* aptr1 = data + (size_t)(row0 + r1) * KD + c1;
    const int wi0 = idx[col0 + r0];
    const int wi1 = idx[col0 + r1];
    const float* bptr0 = weight + (size_t)wi0 * KD + c0;
    const float* bptr1 = weight + (size_t)wi1 * KD + c1;

    v8f acc[4][2] = {};

    const int mf = lane & 15;             // row/col within 16-wide fragment
    const int kb = (lane >> 4) * 2;       // K pair selector per ISA A/B layout

    for (int k0 = 0; k0 < KD; k0 += KT) {
        // Stage A and (gathered) B slices into LDS
        const float4 av0 = *(const float4*)(aptr0 + k0);
        const float4 av1 = *(const float4*)(aptr1 + k0);
        const float4 bv0 = *(const float4*)(bptr0 + k0);
        const float4 bv1 = *(const float4*)(bptr1 + k0);
        *(float4*)&As[r0 * LPAD + c0] = av0;
        *(float4*)&As[r1 * LPAD + c1] = av1;
        *(float4*)&Bs[r0 * LPAD + c0] = bv0;
        *(float4*)&Bs[r1 * LPAD + c1] = bv1;
        __syncthreads();

#pragma unroll
        for (int ks = 0; ks < KT; ks += 4) {
            v2f af[4], bf[2];
#pragma unroll
            for (int sm = 0; sm < 4; ++sm)
                af[sm] = *(const v2f*)&As[(wm * 64 + sm * 16 + mf) * LPAD + ks + kb];
#pragma unroll
            for (int sn = 0; sn < 2; ++sn)
                bf[sn] = *(const v2f*)&Bs[(wn * 32 + sn * 16 + mf) * LPAD + ks + kb];
#pragma unroll
            for (int sm = 0; sm < 4; ++sm)
#pragma unroll
                for (int sn = 0; sn < 2; ++sn)
                    acc[sm][sn] = __builtin_amdgcn_wmma_f32_16x16x4_f32(
                        /*neg_a=*/false, af[sm],
                        /*neg_b=*/false, bf[sn],
                        /*c_mod=*/(short)0, acc[sm][sn],
                        /*reuse_a=*/false, /*reuse_b=*/false);
        }
        __syncthreads();
    }

    // Epilogue: add gathered bias, store.
    // C/D layout: VGPR r holds M = r + (lane>>4)*8, N = lane&15.
    const int mh = (lane >> 4) * 8;
#pragma unroll
    for (int sn = 0; sn < 2; ++sn) {
        const int nG = col0 + wn * 32 + sn * 16 + mf;
        const float bias_v = bsel[nG];
#pragma unroll
        for (int sm = 0; sm < 4; ++sm) {
            const int mbase = row0 + wm * 64 + sm * 16 + mh;
#pragma unroll
            for (int r = 0; r < 8; ++r) {
                out[(size_t)(mbase + r) * N + nG] = acc[sm][sn][r] + bias_v;
            }
        }
    }
}

// ---------------------------------------------------------------------------
extern "C" void kernel_launch(void* const* d_in, const int* in_sizes, int n_in,
                              void* d_out, int out_size, void* d_ws, size_t ws_size,
                              hipStream_t stream) {
    const float* data   = (const float*)d_in[0];
    const float* weight = (const float*)d_in[1];
    const float* bias   = (const float*)d_in[2];
    const int*   mask   = (const int*)d_in[3];
    float*       out    = (float*)d_out;

    const int OUTF = in_sizes[2];               // 16384
    const int KD   = in_sizes[1] / OUTF;        // 4096
    const int M    = in_sizes[0] / KD;          // 8192
    const int N    = out_size / M;              // 8192 (= K kept rows)

    int*   idx  = (int*)d_ws;
    float* bsel = (float*)((char*)d_ws + (size_t)N * sizeof(int));

    compact_mask_kernel<<<1, 1024, 0, stream>>>(mask, bias, idx, bsel, OUTF);

    dim3 grid(N / BN, M / BM);
    gemm_sel_kernel<<<grid, 256, 0, stream>>>(data, weight, idx, bsel, out, M, N, KD);
}